// Router_498216206778
// MI455X (gfx1250) — compile-verified
//
#include <hip/hip_runtime.h>
#include <hip/hip_bf16.h>

// MoE router for MI455X (gfx1250, wave32).
// logits = x[N,2048] @ W[64,2048]^T via V_WMMA_F32_16X16X4_F32 (fp32-exact),
// fused softmax/argmax/aux-loss epilogue. HBM-bound: ~128MB of x @ 23.3TB/s.

#define D_MODEL        2048
#define NUM_E          64
#define TOK_PER_BLOCK  64
#define TOK_PER_WAVE   16
#define KC             64     // K chunk staged in LDS
#define XS             68     // padded LDS row stride in words (68%64=4 -> conflict-free b64 reads)

typedef __attribute__((ext_vector_type(2))) float v2f;
typedef __attribute__((ext_vector_type(8))) float v8f;

__global__ void router_init(int* __restrict__ ws_words, int nwords) {
    int t = blockIdx.x * blockDim.x + threadIdx.x;
    if (t < nwords) ws_words[t] = 0;
}

__global__ __launch_bounds__(128) void router_main(
    const float* __restrict__ x, const float* __restrict__ Wg,
    float* __restrict__ out, int* __restrict__ g_cnt,
    float* __restrict__ g_pr, float* __restrict__ g_z, int N)
{
    __shared__ float xs[TOK_PER_BLOCK * XS];   // x tile; reused for logits after K loop
    __shared__ float wl[NUM_E * XS];           // W tile
    __shared__ float lse_s[TOK_PER_BLOCK];
    __shared__ float pr_s[NUM_E];
    __shared__ int   cnt_s[NUM_E];
    __shared__ float z_s;

    const int tid      = threadIdx.x;
    const int lane     = tid & 31;
    const int wave     = tid >> 5;
    const int row_base = blockIdx.x * TOK_PER_BLOCK;

    if (tid < NUM_E) { pr_s[tid] = 0.0f; cnt_s[tid] = 0; }
    if (tid == 0)    { z_s = 0.0f; }

    v8f acc0 = {}, acc1 = {}, acc2 = {}, acc3 = {};

    const int m    = lane & 15;        // WMMA row (A) / col (B) within 16
    const int koff = (lane >> 4) * 2;  // lanes 16-31 hold K+2,K+3

    for (int k0 = 0; k0 < D_MODEL; k0 += KC) {
        __syncthreads();
        // Stage x (64x64 f32) and W (64x64 f32): float4, 16 lanes = 256B contiguous per row.
        #pragma unroll
        for (int i = 0; i < 8; ++i) {
            int q  = tid + i * 128;        // 1024 float4 per tile
            int r  = q >> 4;
            int c4 = (q & 15) * 4;
            float4 vx = *(const float4*)&x [(size_t)(row_base + r) * D_MODEL + k0 + c4];
            *(float4*)&xs[r * XS + c4] = vx;
            float4 vw = *(const float4*)&Wg[(size_t)r * D_MODEL + k0 + c4];
            *(float4*)&wl[r * XS + c4] = vw;
        }
        __syncthreads();

        #pragma unroll 4
        for (int kk = 0; kk < KC; kk += 4) {
            v2f a  = *(const v2f*)&xs[(wave * TOK_PER_WAVE + m) * XS + kk + koff];
            v2f b0 = *(const v2f*)&wl[( 0 + m) * XS + kk + koff];
            v2f b1 = *(const v2f*)&wl[(16 + m) * XS + kk + koff];
            v2f b2 = *(const v2f*)&wl[(32 + m) * XS + kk + koff];
            v2f b3 = *(const v2f*)&wl[(48 + m) * XS + kk + koff];
            acc0 = __builtin_amdgcn_wmma_f32_16x16x4_f32(false, a, false, b0, (short)0, acc0, false, false);
            acc1 = __builtin_amdgcn_wmma_f32_16x16x4_f32(false, a, false, b1, (short)0, acc1, false, false);
            acc2 = __builtin_amdgcn_wmma_f32_16x16x4_f32(false, a, false, b2, (short)0, acc2, false, false);
            acc3 = __builtin_amdgcn_wmma_f32_16x16x4_f32(false, a, false, b3, (short)0, acc3, false, false);
        }
    }

    __syncthreads();                 // everyone done reading xs -> reuse as logits buffer
    float* lg = xs;                  // lg[token][expert], stride XS
    {
        // C/D layout: VGPR i -> M=i (lanes 0-15) or M=i+8 (lanes 16-31); lane%16 = N
        const int rbase = wave * TOK_PER_WAVE + ((lane >> 4) << 3);
        const int col   = lane & 15;
        #pragma unroll
        for (int i = 0; i < 8; ++i) {
            lg[(rbase + i) * XS +  0 + col] = acc0[i];
            lg[(rbase + i) * XS + 16 + col] = acc1[i];
            lg[(rbase + i) * XS + 32 + col] = acc2[i];
            lg[(rbase + i) * XS + 48 + col] = acc3[i];
        }
    }
    __syncthreads();

    // Per-token softmax stats: threads 0..63 each own one token.
    if (tid < TOK_PER_BLOCK) {
        const float* lr = &lg[tid * XS];
        float mx = lr[0]; int am = 0;
        #pragma unroll 8
        for (int e = 1; e < NUM_E; ++e) { float v = lr[e]; if (v > mx) { mx = v; am = e; } }
        float s = 0.0f;
        #pragma unroll 8
        for (int e = 0; e < NUM_E; ++e) s += __expf(lr[e] - mx);
        float lse = mx + __logf(s);
        lse_s[tid] = lse;
        int tok = row_base + tid;
        out[tok]     = (float)am;            // expert_index (as float)
        out[N + tok] = __expf(mx - lse);     // expert_prob = probs[argmax]
        atomicAdd(&cnt_s[am], 1);
        atomicAdd(&z_s, lse * lse);
    }
    __syncthreads();

    // Per-expert prob sums: thread handles expert tid&63, half of the tokens.
    {
        int e    = tid & 63;
        int half = tid >> 6;
        float a = 0.0f;
        #pragma unroll 8
        for (int t = half * 32; t < half * 32 + 32; ++t)
            a += __expf(lg[t * XS + e] - lse_s[t]);
        atomicAdd(&pr_s[e], a);
    }
    __syncthreads();

    if (tid < NUM_E) {
        atomicAdd(&g_pr[tid],  pr_s[tid]);
        atomicAdd(&g_cnt[tid], cnt_s[tid]);
    }
    if (tid == 0) atomicAdd(g_z, z_s);
}

__global__ void router_finalize(float* __restrict__ out, const int* __restrict__ g_cnt,
                                const float* __restrict__ g_pr, const float* __restrict__ g_z,
                                int N)
{
    __shared__ float fp[NUM_E];
    int e = threadIdx.x;
    float c = (float)g_cnt[e];
    out[2 * N + e] = c;                                        // counts (as float)
    fp[e] = (c / (float)N) * (g_pr[e] / (float)N);             // f_i * p_i
    __syncthreads();
    if (e == 0) {
        float dot = 0.0f;
        for (int i = 0; i < NUM_E; ++i) dot += fp[i];
        float z_loss = g_z[0] / (float)N;
        float aux = 1.0e-2f * (float)NUM_E * dot + 1.0e-3f * z_loss;
        int cap = (N + NUM_E - 1) / NUM_E;
        if (cap < 4) cap = 4;
        out[2 * N + NUM_E]     = (float)cap;                   // capacity
        out[2 * N + NUM_E + 1] = aux;                          // aux_loss
    }
}

extern "C" void kernel_launch(void* const* d_in, const int* in_sizes, int n_in,
                              void* d_out, int out_size, void* d_ws, size_t ws_size,
                              hipStream_t stream) {
    const float* x  = (const float*)d_in[0];
    const float* Wg = (const float*)d_in[1];
    const int N = in_sizes[0] / D_MODEL;       // 16384

    int*   g_cnt = (int*)d_ws;                 // [64]
    float* g_pr  = (float*)d_ws + NUM_E;       // [64]
    float* g_z   = (float*)d_ws + 2 * NUM_E;   // [1]

    router_init<<<1, 256, 0, stream>>>((int*)d_ws, 2 * NUM_E + 1);
    router_main<<<N / TOK_PER_BLOCK, 128, 0, stream>>>(
        x, Wg, (float*)d_out, g_cnt, g_pr, g_z, N);
    router_finalize<<<1, NUM_E, 0, stream>>>((float*)d_out, g_cnt, g_pr, g_z, N);
}